// AdaDecomp_63144609185943
// MI455X (gfx1250) — compile-verified
//
#include <hip/hip_runtime.h>

// ============================================================================
// 3-level Haar DWT + per-band reconstruction  (B=32, T=4096, N=64, fp32)
//
// The whole pipeline is a banded linear operator along T, identical for every
// (b, n) channel. Interior structure: out rows [16m, 16m+16) depend only on
// input rows [16m-7, 16m+16] (24 taps), shift-invariant with period 16.
// => per tile:  OUT_o(16x16) = W_o(16x24) * X(24x16)   for o = 0..3
// computed with 6 chained V_WMMA_F32_16X16X4_F32 per output (24 WMMA / wave).
// Boundary rows (t<16, t>=4080), where symmetric padding & odd intermediate
// lengths (2049/1026/514) break shift invariance, are recomputed exactly by a
// scalar fixup kernel.  Memory-bound: ~176 MB traffic => ~7.5 us @ 23.3 TB/s.
// ============================================================================

typedef float v2f __attribute__((ext_vector_type(2)));
typedef float v8f __attribute__((ext_vector_type(8)));

#define T_LEN 4096
#define N_CH  64

// ---------------------------------------------------------------------------
// Exact reference recurrences (including boundary branches) for LEVELS=3,
// 2-tap filters h=[h0,h1], g=[g0,g1], input length 4096.
//   c1 len 2049, c2 len 1026, c3 len 514.
// Synthesis (filters reversed, lhs_dilation=2):
//   r[m] even: h1*lo[m/2] + g1*hi[m/2] ; odd: h0*lo[(m+1)/2] + g0*hi[(m+1)/2]
// Output selector o: 0=x_l (c3 path), 1=xh(level3), 2=xh(level2), 3=xh(level1)
// ---------------------------------------------------------------------------
template <typename F>
struct Pipe {
  F x;
  float h0, h1, g0, g1;

  __device__ float c1(int i) const {
    if (i == 0)    return (h0 + h1) * x(0);
    if (i == 2048) return (h0 + h1) * x(4095);
    return h0 * x(2 * i - 1) + h1 * x(2 * i);
  }
  __device__ float d1(int i) const {
    if (i == 0)    return (g0 + g1) * x(0);
    if (i == 2048) return (g0 + g1) * x(4095);
    return g0 * x(2 * i - 1) + g1 * x(2 * i);
  }
  __device__ float c2(int j) const {
    if (j == 0)    return (h0 + h1) * c1(0);
    if (j == 1025) return h0 * c1(2048) + h1 * c1(2047); // right pad = 2 (odd len)
    return h0 * c1(2 * j - 1) + h1 * c1(2 * j);
  }
  __device__ float d2(int j) const {
    if (j == 0)    return (g0 + g1) * c1(0);
    if (j == 1025) return g0 * c1(2048) + g1 * c1(2047);
    return g0 * c1(2 * j - 1) + g1 * c1(2 * j);
  }
  __device__ float c3(int k) const {
    if (k == 0)   return (h0 + h1) * c2(0);
    if (k == 513) return (h0 + h1) * c2(1025);
    return h0 * c2(2 * k - 1) + h1 * c2(2 * k);
  }
  __device__ float d3(int k) const {
    if (k == 0)   return (g0 + g1) * c2(0);
    if (k == 513) return (g0 + g1) * c2(1025);
    return g0 * c2(2 * k - 1) + g1 * c2(2 * k);
  }
  __device__ float r2f(int o, int m) const {
    int idx = (m + (m & 1)) >> 1;                    // ceil(m/2)
    float lo = (o == 0) ? c3(idx) : 0.0f;
    float hi = (o == 1) ? d3(idx) : 0.0f;
    return (m & 1) ? (h0 * lo + g0 * hi) : (h1 * lo + g1 * hi);
  }
  __device__ float r1f(int o, int p) const {
    int idx = (p + (p & 1)) >> 1;
    float lo = r2f(o, idx);
    float hi = (o == 2) ? d2(idx) : 0.0f;
    return (p & 1) ? (h0 * lo + g0 * hi) : (h1 * lo + g1 * hi);
  }
  __device__ float outv(int o, int t) const {
    int idx = (t + (t & 1)) >> 1;                    // trim of r1 never indexed
    float lo = r1f(o, idx);
    float hi = (o == 3) ? d1(idx) : 0.0f;
    return (t & 1) ? (h0 * lo + g0 * hi) : (h1 * lo + g1 * hi);
  }
};

struct Impulse {
  int g;
  __device__ float operator()(int i) const { return (i == g) ? 1.0f : 0.0f; }
};

struct GlobalX {
  const float* p;  // X[b, 0, n], stride N_CH along T
  __device__ float operator()(int i) const { return p[(size_t)i * N_CH]; }
};

// ---------------------------------------------------------------------------
// Build interior band W_o[16 rows][24 taps] by impulse response at a reference
// interior tile (m_ref=4: taps cover x[57..80], outputs t=64..79).
// Stored directly in WMMA A-matrix layout (f32 16x16x4):
//   lane L holds M=L%16 ; VGPR0 <-> K = 2*(L/16)+0 ; VGPR1 <-> K = 2*(L/16)+1
// Buffer index: W[(((o*6+q)*2 + j) << 5) + lane], q = K-chunk (0..5), j = VGPR.
// ---------------------------------------------------------------------------
__global__ void build_weights(const float* __restrict__ h,
                              const float* __restrict__ g,
                              float* __restrict__ W) {
  int idx = blockIdx.x * blockDim.x + threadIdx.x;
  if (idx >= 4 * 6 * 2 * 32) return;
  int lane = idx & 31;
  int j    = (idx >> 5) & 1;
  int q    = (idx >> 6) % 6;
  int o    = idx / 384;
  int r    = lane & 15;
  int d    = 4 * q + 2 * (lane >> 4) + j;            // tap index 0..23
  Pipe<Impulse> P{Impulse{57 + d}, h[0], h[1], g[0], g[1]};
  W[idx] = P.outv(o, 64 + r);
}

// ---------------------------------------------------------------------------
// Main tile kernel: 1 wave = 16 t-rows x 16 channels, all 4 outputs.
// Grid: (254 tiles m=1..254, B) x block 128 (4 waves = 4 channel tiles).
// B-matrix layout (f32 4x16): lane L holds N=L%16 ;
//   VGPR0 <-> K-row = 2*(L/16)+0 ; VGPR1 <-> K-row = 2*(L/16)+1.
// ---------------------------------------------------------------------------
__global__ void __launch_bounds__(128)
wavelet_wmma(const float* __restrict__ X, const float* __restrict__ W,
             float* __restrict__ out, int B) {
  const int lane = threadIdx.x & 31;
  const int ct   = threadIdx.x >> 5;                 // channel tile 0..3
  const int m    = blockIdx.x + 1;                   // 1..254
  const int b    = blockIdx.y;
  const int half = lane >> 4;
  const int ln   = lane & 15;
  const int c0   = ct * 16;

  // Load all A tiles (4 outputs x 6 K-chunks x 2 VGPRs) — coalesced, L2 hot.
  float a[4][6][2];
#pragma unroll
  for (int o = 0; o < 4; ++o)
#pragma unroll
    for (int q = 0; q < 6; ++q)
#pragma unroll
      for (int j = 0; j < 2; ++j)
        a[o][q][j] = W[(((o * 6 + q) * 2 + j) << 5) + lane];

  const float* xb   = X + (size_t)b * T_LEN * N_CH;
  const int    base = 16 * m - 7;                    // tap window start

  // Warm L2 for the next tile's rows (gfx1250 global_prefetch_b8).
  __builtin_prefetch(xb + (size_t)(base + 24) * N_CH + c0 + ln, 0, 1);

  v8f acc0 = {}, acc1 = {}, acc2 = {}, acc3 = {};
#pragma unroll
  for (int q = 0; q < 6; ++q) {
    const float* px = xb + (size_t)(base + 4 * q + 2 * half) * N_CH + c0 + ln;
    v2f bv;
    bv.x = px[0];        // K-row 2*half
    bv.y = px[N_CH];     // K-row 2*half+1
    v2f av;
    av.x = a[0][q][0]; av.y = a[0][q][1];
    acc0 = __builtin_amdgcn_wmma_f32_16x16x4_f32(false, av, false, bv,
                                                 (short)0, acc0, false, false);
    av.x = a[1][q][0]; av.y = a[1][q][1];
    acc1 = __builtin_amdgcn_wmma_f32_16x16x4_f32(false, av, false, bv,
                                                 (short)0, acc1, false, false);
    av.x = a[2][q][0]; av.y = a[2][q][1];
    acc2 = __builtin_amdgcn_wmma_f32_16x16x4_f32(false, av, false, bv,
                                                 (short)0, acc2, false, false);
    av.x = a[3][q][0]; av.y = a[3][q][1];
    acc3 = __builtin_amdgcn_wmma_f32_16x16x4_f32(false, av, false, bv,
                                                 (short)0, acc3, false, false);
  }

  // D layout: VGPR v, lanes 0-15 -> M=v ; lanes 16-31 -> M=v+8.
  const size_t osz = (size_t)B * T_LEN * N_CH;
  size_t ob = ((size_t)b * T_LEN + 16 * m + 8 * half) * N_CH + c0 + ln;
  v8f accs[4] = {acc0, acc1, acc2, acc3};
#pragma unroll
  for (int o = 0; o < 4; ++o)
#pragma unroll
    for (int v = 0; v < 8; ++v)
      out[o * osz + ob + (size_t)v * N_CH] = accs[o][v];
}

// ---------------------------------------------------------------------------
// Exact recomputation of boundary rows t in [0,16) U [4080,4096).
// ---------------------------------------------------------------------------
__global__ void edge_fixup(const float* __restrict__ X,
                           const float* __restrict__ h,
                           const float* __restrict__ g,
                           float* __restrict__ out, int B) {
  int idx = blockIdx.x * blockDim.x + threadIdx.x;
  if (idx >= B * 32 * N_CH) return;
  int n  = idx & (N_CH - 1);
  int tt = (idx >> 6) & 31;
  int b  = idx >> 11;
  int t  = (tt < 16) ? tt : (T_LEN - 16 + (tt - 16));
  Pipe<GlobalX> P{GlobalX{X + (size_t)b * T_LEN * N_CH + n},
                  h[0], h[1], g[0], g[1]};
  const size_t osz = (size_t)B * T_LEN * N_CH;
  size_t ob = ((size_t)b * T_LEN + t) * N_CH + n;
  for (int o = 0; o < 4; ++o) out[o * osz + ob] = P.outv(o, t);
}

// ---------------------------------------------------------------------------
extern "C" void kernel_launch(void* const* d_in, const int* in_sizes, int n_in,
                              void* d_out, int out_size, void* d_ws,
                              size_t ws_size, hipStream_t stream) {
  const float* X = (const float*)d_in[0];
  const float* h = (const float*)d_in[1];   // h_dec (2 taps)
  const float* g = (const float*)d_in[2];   // g_dec (2 taps)
  float* out = (float*)d_out;               // [x_l | xh_l3 | xh_l2 | xh_l1]
  float* W   = (float*)d_ws;                // 1536 floats = 6 KB
  int B = in_sizes[0] / (T_LEN * N_CH);     // 32

  build_weights<<<6, 256, 0, stream>>>(h, g, W);
  wavelet_wmma<<<dim3(254, B), 128, 0, stream>>>(X, W, out, B);
  edge_fixup<<<(B * 32 * N_CH + 255) / 256, 256, 0, stream>>>(X, h, g, out, B);
}